// MODEL_77309411328168
// MI455X (gfx1250) — compile-verified
//
#include <hip/hip_runtime.h>

#define USER_NUM 100000
#define ITEM_NUM 50000
#define N_NODES  (USER_NUM + ITEM_NUM)
#define NNZ      4800000
#define HD       128
#define OUTD     384
#define SLOPE    0.2f
#define EPSN     1e-12f

typedef float v2f __attribute__((ext_vector_type(2)));
typedef float v8f __attribute__((ext_vector_type(8)));

// ---------------------------------------------------------------------------
// Zero fill (float4 vectorized)
// ---------------------------------------------------------------------------
__global__ void zero_f4_kernel(float4* __restrict__ p, long n4) {
    long i = (long)blockIdx.x * blockDim.x + threadIdx.x;
    long stride = (long)gridDim.x * blockDim.x;
    float4 z; z.x = 0.f; z.y = 0.f; z.z = 0.f; z.w = 0.f;
    for (; i < n4; i += stride) p[i] = z;
}

// ---------------------------------------------------------------------------
// Copy raw embeddings into output columns [0,128)
// one wave per row, float4 per lane
// ---------------------------------------------------------------------------
__global__ __launch_bounds__(256)
void copy_emb_kernel(const float* __restrict__ uemb,
                     const float* __restrict__ iemb,
                     float* __restrict__ out) {
    int lane = threadIdx.x & 31;
    long wid = ((long)blockIdx.x * blockDim.x + threadIdx.x) >> 5;
    long nw  = ((long)gridDim.x * blockDim.x) >> 5;
    for (long r = wid; r < N_NODES; r += nw) {
        const float* src = (r < USER_NUM) ? (uemb + (size_t)r * HD)
                                          : (iemb + (size_t)(r - USER_NUM) * HD);
        float4 v = ((const float4*)src)[lane];
        ((float4*)(out + (size_t)r * OUTD))[lane] = v;
    }
}

// ---------------------------------------------------------------------------
// Dense GEMM  Y[16-row strip] = X_strip @ W   via V_WMMA_F32_16X16X4_F32
// Block = 256 threads = 8 waves; wave w owns output cols [16w,16w+16).
// A strip (16x128) staged in LDS via GLOBAL_LOAD_ASYNC_TO_LDS_B128
// (ASYNCcnt-tracked, no VGPR round trip), shared by all 8 waves.
// Fragment mapping per CDNA5 ISA 7.12.2:
//   A 16x4 f32 : lane m=lane%16, vgpr0/1 hold K = 2*(lane/16)+{0,1}
//   C 16x16 f32: lanes 0-15 -> M=j, lanes 16-31 -> M=8+j, N=lane%16
// ---------------------------------------------------------------------------
__global__ __launch_bounds__(256)
void gemm_wmma_kernel(const float* __restrict__ X,
                      const float* __restrict__ W,
                      float* __restrict__ Y) {
    __shared__ float As[16 * HD];
    const int tid  = threadIdx.x;
    const int row0 = blockIdx.x * 16;

    // stage 16x128 A strip (8KB): 256 threads x 2 async b128 copies
    {
        const float* g0 = X + (size_t)row0 * HD + (size_t)tid * 4;
        const float* g1 = g0 + 1024;                       // +4096 bytes
        unsigned l0 = (unsigned)(size_t)(&As[0]) + (unsigned)tid * 16u;
        unsigned l1 = l0 + 4096u;
        asm volatile("global_load_async_to_lds_b128 %0, %1, off"
                     :: "v"(l0), "v"(g0) : "memory");
        asm volatile("global_load_async_to_lds_b128 %0, %1, off"
                     :: "v"(l1), "v"(g1) : "memory");
        asm volatile("s_wait_asynccnt 0x0" ::: "memory");
    }
    __syncthreads();

    const int wave  = tid >> 5;
    const int lane  = tid & 31;
    const int m     = lane & 15;          // row within tile (A) / col (B,C)
    const int khalf = (lane >> 4) * 2;    // 0 or 2
    const int n     = wave * 16 + m;      // output column

    v8f c = {};
    #pragma unroll 4
    for (int kk = 0; kk < HD; kk += 4) {
        v2f a, b;
        a.x = As[m * HD + kk + khalf];
        a.y = As[m * HD + kk + khalf + 1];
        b.x = W[(size_t)(kk + khalf)     * HD + n];
        b.y = W[(size_t)(kk + khalf + 1) * HD + n];
        c = __builtin_amdgcn_wmma_f32_16x16x4_f32(
                false, a, false, b, (short)0, c, false, false);
    }

    const int mrow = row0 + (lane >> 4) * 8;
    #pragma unroll
    for (int j = 0; j < 8; ++j) {
        Y[(size_t)(mrow + j) * HD + n] = c[j];
    }
}

// ---------------------------------------------------------------------------
// SpMM scatter: accum[rows[e]] += vals[e] * feat[cols[e]]
// one wave per edge (grid-stride); 512B coalesced gather, 4 f32 atomics/lane
// feat (76.8MB) is L2-resident on MI455X (192MB L2) -> gathers are L2 hits;
// scatter side rides the L2 atomic units (no-return, STOREcnt-tracked).
// ---------------------------------------------------------------------------
__global__ __launch_bounds__(256)
void spmm_scatter_kernel(const int* __restrict__ rows,
                         const int* __restrict__ cols,
                         const float* __restrict__ vals,
                         const float* __restrict__ feat,
                         float* __restrict__ accum) {
    int lane = threadIdx.x & 31;
    long wid = ((long)blockIdx.x * blockDim.x + threadIdx.x) >> 5;
    long nw  = ((long)gridDim.x * blockDim.x) >> 5;
    for (long e = wid; e < NNZ; e += nw) {
        int   r = rows[e];
        int   c = cols[e];
        float v = vals[e];
        float4 f = ((const float4*)(feat + (size_t)c * HD))[lane];
        float* dst = accum + (size_t)r * HD + (size_t)lane * 4;
        atomicAdd(dst + 0, v * f.x);
        atomicAdd(dst + 1, v * f.y);
        atomicAdd(dst + 2, v * f.z);
        atomicAdd(dst + 3, v * f.w);
    }
}

// ---------------------------------------------------------------------------
// LeakyReLU (in place, feeds next layer) + row L2-normalize -> out[:,col_off]
// one wave per row; wave32 shuffle reduction
// ---------------------------------------------------------------------------
__global__ __launch_bounds__(256)
void leaky_norm_kernel(float* __restrict__ accum,
                       float* __restrict__ out,
                       int col_off) {
    int lane = threadIdx.x & 31;
    long wid = ((long)blockIdx.x * blockDim.x + threadIdx.x) >> 5;
    long nw  = ((long)gridDim.x * blockDim.x) >> 5;
    for (long r = wid; r < N_NODES; r += nw) {
        float4* rowp = (float4*)(accum + (size_t)r * HD);
        float4 x = rowp[lane];
        x.x = (x.x > 0.f) ? x.x : SLOPE * x.x;
        x.y = (x.y > 0.f) ? x.y : SLOPE * x.y;
        x.z = (x.z > 0.f) ? x.z : SLOPE * x.z;
        x.w = (x.w > 0.f) ? x.w : SLOPE * x.w;
        rowp[lane] = x;  // layer1 consumes the UNnormalized activation

        float ss = x.x * x.x + x.y * x.y + x.z * x.z + x.w * x.w;
        #pragma unroll
        for (int off = 16; off > 0; off >>= 1)
            ss += __shfl_xor(ss, off, 32);
        float inv = 1.0f / fmaxf(sqrtf(ss), EPSN);

        float4 y;
        y.x = x.x * inv; y.y = x.y * inv; y.z = x.z * inv; y.w = x.w * inv;
        ((float4*)(out + (size_t)r * OUTD + col_off))[lane] = y;
    }
}

// ---------------------------------------------------------------------------
extern "C" void kernel_launch(void* const* d_in, const int* in_sizes, int n_in,
                              void* d_out, int out_size, void* d_ws, size_t ws_size,
                              hipStream_t stream) {
    const int*   rows     = (const int*)  d_in[0];
    const int*   cols     = (const int*)  d_in[1];
    const float* vals     = (const float*)d_in[2];
    const float* user_emb = (const float*)d_in[3];
    const float* item_emb = (const float*)d_in[4];
    const float* u_w0     = (const float*)d_in[5];
    const float* v_w0     = (const float*)d_in[6];
    const float* u_w1     = (const float*)d_in[7];
    const float* v_w1     = (const float*)d_in[8];
    float* out = (float*)d_out;

    float* feat  = (float*)d_ws;                       // [N_NODES, 128]
    float* accum = feat + (size_t)N_NODES * HD;        // [N_NODES, 128]
    const long n4 = (long)N_NODES * HD / 4;

    // level-0 raw embeddings -> out[:, 0:128)
    copy_emb_kernel<<<1024, 256, 0, stream>>>(user_emb, item_emb, out);

    // ---- layer 0 ----
    zero_f4_kernel<<<2048, 256, 0, stream>>>((float4*)accum, n4);
    gemm_wmma_kernel<<<USER_NUM / 16, 256, 0, stream>>>(user_emb, u_w0, feat);
    gemm_wmma_kernel<<<ITEM_NUM / 16, 256, 0, stream>>>(item_emb, v_w0,
                                                        feat + (size_t)USER_NUM * HD);
    spmm_scatter_kernel<<<4096, 256, 0, stream>>>(rows, cols, vals, feat, accum);
    leaky_norm_kernel<<<2048, 256, 0, stream>>>(accum, out, HD);   // out[:,128:256)

    // ---- layer 1 (consumes unnormalized leaky output in `accum`) ----
    gemm_wmma_kernel<<<USER_NUM / 16, 256, 0, stream>>>(accum, u_w1, feat);
    gemm_wmma_kernel<<<ITEM_NUM / 16, 256, 0, stream>>>(accum + (size_t)USER_NUM * HD,
                                                        v_w1,
                                                        feat + (size_t)USER_NUM * HD);
    zero_f4_kernel<<<2048, 256, 0, stream>>>((float4*)accum, n4);
    spmm_scatter_kernel<<<4096, 256, 0, stream>>>(rows, cols, vals, feat, accum);
    leaky_norm_kernel<<<2048, 256, 0, stream>>>(accum, out, 2 * HD); // out[:,256:384)
}